// SMPLX_32366873543174
// MI455X (gfx1250) — compile-verified
//
#include <hip/hip_runtime.h>

typedef float v2f __attribute__((ext_vector_type(2)));
typedef float v8f __attribute__((ext_vector_type(8)));

#define B_    256
#define V_    10475
#define J_    55
#define JP_   56       // J padded (rel has a zero row at j=55)
#define NB_   10
#define P_    486      // (J-1)*9
#define N3_   31425    // V*3
#define KP_   488      // P padded to multiple of 4 (pf zero-padded)
#define NGF_  491      // full 64-wide N groups: 491*64 = 31424 (leftover col n=31424)
#define VT_   655      // ceil(V/16)

// ---------------------------------------------------------------------------
// 1) v_shaped[b][v][k] = v_template[v][k] + sum_l shapedirs[v][k][l]*betas[b][l]
// ---------------------------------------------------------------------------
__global__ __launch_bounds__(256) void k_shape(
    const float* __restrict__ betas, const float* __restrict__ v_template,
    const float* __restrict__ shapedirs, float* __restrict__ vshaped) {
  int idx = blockIdx.x * blockDim.x + threadIdx.x;
  if (idx >= B_ * V_) return;
  int b = idx / V_;
  int v = idx - b * V_;
  const float* be = betas + b * NB_;
  const float* sd = shapedirs + (size_t)v * 3 * NB_;
  float bb[NB_];
#pragma unroll
  for (int l = 0; l < NB_; ++l) bb[l] = be[l];
#pragma unroll
  for (int k = 0; k < 3; ++k) {
    float acc = v_template[v * 3 + k];
#pragma unroll
    for (int l = 0; l < NB_; ++l) acc += sd[k * NB_ + l] * bb[l];
    vshaped[(size_t)b * N3_ + v * 3 + k] = acc;
  }
}

// ---------------------------------------------------------------------------
// 2) Rodrigues: rot_mats[b][j][9], pose_feature[b][KP_] (R - I for j>=1, pad=0)
// ---------------------------------------------------------------------------
__global__ __launch_bounds__(256) void k_rodrigues(
    const float* __restrict__ pose, float* __restrict__ rotm,
    float* __restrict__ pf) {
  int idx = blockIdx.x * blockDim.x + threadIdx.x;
  if (idx >= B_ * J_) return;
  int b = idx / J_;
  int j = idx - b * J_;
  float x = pose[(b * J_ + j) * 3 + 0];
  float y = pose[(b * J_ + j) * 3 + 1];
  float z = pose[(b * J_ + j) * 3 + 2];
  float ax = x + 1e-8f, ay = y + 1e-8f, az = z + 1e-8f;
  float angle = sqrtf(ax * ax + ay * ay + az * az);
  float inv = 1.0f / angle;
  float rx = x * inv, ry = y * inv, rz = z * inv;
  float s = sinf(angle), c = cosf(angle), t = 1.0f - c;
  float R[9];
  R[0] = c + t * rx * rx;       R[1] = -s * rz + t * rx * ry;  R[2] = s * ry + t * rx * rz;
  R[3] = s * rz + t * rx * ry;  R[4] = c + t * ry * ry;        R[5] = -s * rx + t * ry * rz;
  R[6] = -s * ry + t * rx * rz; R[7] = s * rx + t * ry * rz;   R[8] = c + t * rz * rz;
#pragma unroll
  for (int q = 0; q < 9; ++q) rotm[((size_t)b * J_ + j) * 9 + q] = R[q];
  if (j > 0) {
#pragma unroll
    for (int q = 0; q < 9; ++q)
      pf[(size_t)b * KP_ + (j - 1) * 9 + q] = R[q] - ((q % 4 == 0) ? 1.0f : 0.0f);
  } else {
    pf[(size_t)b * KP_ + 486] = 0.0f;   // zero the K padding
    pf[(size_t)b * KP_ + 487] = 0.0f;
  }
}

// ---------------------------------------------------------------------------
// 3) Jloc[b][j][k] = sum_v J_regressor[j][v] * vshaped[b][v][k]
// ---------------------------------------------------------------------------
__global__ __launch_bounds__(256) void k_jloc(
    const float* __restrict__ Jr, const float* __restrict__ vshaped,
    float* __restrict__ jloc) {
  int wid = blockIdx.x * (blockDim.x >> 5) + (threadIdx.x >> 5);
  if (wid >= B_ * J_) return;
  int lane = threadIdx.x & 31;
  int b = wid / J_;
  int j = wid - b * J_;
  const float* w  = Jr + (size_t)j * V_;
  const float* vs = vshaped + (size_t)b * N3_;
  float a0 = 0.f, a1 = 0.f, a2 = 0.f;
  for (int v = lane; v < V_; v += 32) {
    float ww = w[v];
    a0 += ww * vs[v * 3 + 0];
    a1 += ww * vs[v * 3 + 1];
    a2 += ww * vs[v * 3 + 2];
  }
#pragma unroll
  for (int off = 16; off > 0; off >>= 1) {
    a0 += __shfl_down(a0, off);
    a1 += __shfl_down(a1, off);
    a2 += __shfl_down(a2, off);
  }
  if (lane == 0) {
    jloc[((size_t)b * J_ + j) * 3 + 0] = a0;
    jloc[((size_t)b * J_ + j) * 3 + 1] = a1;
    jloc[((size_t)b * J_ + j) * 3 + 2] = a2;
  }
}

// ---------------------------------------------------------------------------
// 4) Kinematic chain: 16 lanes per batch, chain kept in LDS.
//    Emits rel[b][JP_][16] (row 55 zeroed) and posed_joints (into d_out).
// ---------------------------------------------------------------------------
__global__ __launch_bounds__(256) void k_chain(
    const float* __restrict__ rotm, const float* __restrict__ jloc,
    const int* __restrict__ parents, float* __restrict__ rel,
    float* __restrict__ joints) {
  __shared__ float Tch[16][J_][16];   // 56320 B
  __shared__ float tmS[16][16];
  int lb = threadIdx.x >> 4;
  int e  = threadIdx.x & 15;
  int b  = blockIdx.x * 16 + lb;
  int r = e >> 2, c = e & 3;
  for (int i = 0; i < J_; ++i) {
    int p = parents[i];
    float rj = 0.f;
    if (r < 3) {
      rj = jloc[((size_t)b * J_ + i) * 3 + r];
      if (i > 0) rj -= jloc[((size_t)b * J_ + p) * 3 + r];
    }
    float tm;
    if (c < 3) tm = (r < 3) ? rotm[((size_t)b * J_ + i) * 9 + r * 3 + c] : 0.f;
    else       tm = (r < 3) ? rj : 1.f;
    __syncthreads();                 // protect tmS WAR from previous iter
    tmS[lb][e] = tm;
    __syncthreads();
    float Te;
    if (i == 0) {
      Te = tm;
    } else {
      float acc = 0.f;
#pragma unroll
      for (int k = 0; k < 4; ++k) acc += Tch[lb][p][r * 4 + k] * tmS[lb][k * 4 + c];
      Te = acc;
    }
    Tch[lb][i][e] = Te;
    __syncthreads();
    if (c == 3 && r < 3) joints[((size_t)b * J_ + i) * 3 + r] = Te;
    float relv = Te;
    if (c == 3 && r < 3) {
      float sdot = 0.f;
#pragma unroll
      for (int k = 0; k < 3; ++k)
        sdot += Tch[lb][i][r * 4 + k] * jloc[((size_t)b * J_ + i) * 3 + k];
      relv = Te - sdot;
    }
    rel[((size_t)b * JP_ + i) * 16 + e] = relv;
  }
  rel[((size_t)b * JP_ + J_) * 16 + e] = 0.f;   // zero pad row j=55
}

// ---------------------------------------------------------------------------
// 5) Pose GEMM (WMMA f32 16x16x4), guard-free steady state.
//    Each wave: 16(M batches) x 64(N) strip; covers n in [0, 31424).
// ---------------------------------------------------------------------------
__global__ __launch_bounds__(256) void k_pose_gemm(
    const float* __restrict__ pf, const float* __restrict__ pd,
    const float* __restrict__ vshaped, float* __restrict__ vposed) {
  int wid = blockIdx.x * (blockDim.x >> 5) + (threadIdx.x >> 5);
  int mt = wid / NGF_;                // 0..15
  int ng = wid - mt * NGF_;           // 0..490
  int m0 = mt * 16;
  int n0 = ng * 64;
  int lane = threadIdx.x & 31;
  int kp  = (lane >> 4) * 2;          // f32 A/B layout: lanes 16-31 hold K+2
  int col = lane & 15;
  const float* pfrow = pf + (size_t)(m0 + (lane & 15)) * KP_;
  const float* pdc   = pd + n0 + col;

  v8f acc0 = (v8f){0,0,0,0,0,0,0,0};
  v8f acc1 = acc0, acc2 = acc0, acc3 = acc0;

  // main: k = 0..483, all loads provably in bounds (no guards, no branches)
  for (int k0 = 0; k0 < 484; k0 += 4) {
    int ka = k0 + kp, kb = ka + 1;
    v2f a;
    a.x = pfrow[ka];
    a.y = pfrow[kb];
    size_t ra = (size_t)ka * N3_;
    size_t rb = (size_t)kb * N3_;
    v2f b0, b1, b2, b3;
    b0.x = pdc[ra +  0]; b0.y = pdc[rb +  0];
    b1.x = pdc[ra + 16]; b1.y = pdc[rb + 16];
    b2.x = pdc[ra + 32]; b2.y = pdc[rb + 32];
    b3.x = pdc[ra + 48]; b3.y = pdc[rb + 48];
    acc0 = __builtin_amdgcn_wmma_f32_16x16x4_f32(false, a, false, b0, (short)0, acc0, false, false);
    acc1 = __builtin_amdgcn_wmma_f32_16x16x4_f32(false, a, false, b1, (short)0, acc1, false, false);
    acc2 = __builtin_amdgcn_wmma_f32_16x16x4_f32(false, a, false, b2, (short)0, acc2, false, false);
    acc3 = __builtin_amdgcn_wmma_f32_16x16x4_f32(false, a, false, b3, (short)0, acc3, false, false);
  }
  // epilogue: k = 484..487. pf pad is zero, so A=0 for k>=486; B address is
  // clamped to a valid row -> 0 * finite = 0, still branch-free.
  {
    int ka = 484 + kp, kb = ka + 1;
    v2f a;
    a.x = pfrow[ka];                      // 486/487 read zero pad
    a.y = pfrow[kb];
    size_t ra = (size_t)min(ka, P_ - 1) * N3_;
    size_t rb = (size_t)min(kb, P_ - 1) * N3_;
    v2f b0, b1, b2, b3;
    b0.x = pdc[ra +  0]; b0.y = pdc[rb +  0];
    b1.x = pdc[ra + 16]; b1.y = pdc[rb + 16];
    b2.x = pdc[ra + 32]; b2.y = pdc[rb + 32];
    b3.x = pdc[ra + 48]; b3.y = pdc[rb + 48];
    acc0 = __builtin_amdgcn_wmma_f32_16x16x4_f32(false, a, false, b0, (short)0, acc0, false, false);
    acc1 = __builtin_amdgcn_wmma_f32_16x16x4_f32(false, a, false, b1, (short)0, acc1, false, false);
    acc2 = __builtin_amdgcn_wmma_f32_16x16x4_f32(false, a, false, b2, (short)0, acc2, false, false);
    acc3 = __builtin_amdgcn_wmma_f32_16x16x4_f32(false, a, false, b3, (short)0, acc3, false, false);
  }
  int hi = lane >> 4;
  v8f accs[4] = {acc0, acc1, acc2, acc3};
#pragma unroll
  for (int s = 0; s < 4; ++s) {
    int n = n0 + s * 16 + col;            // always < 31424: no guard
#pragma unroll
    for (int r = 0; r < 8; ++r) {
      int brow = m0 + r + hi * 8;         // C/D layout: VGPR r -> M = r / r+8
      vposed[(size_t)brow * N3_ + n] = accs[s][r] + vshaped[(size_t)brow * N3_ + n];
    }
  }
}

// ---------------------------------------------------------------------------
// 5b) leftover column n = 31424 (scalar, 256 threads)
// ---------------------------------------------------------------------------
__global__ __launch_bounds__(256) void k_pose_lastcol(
    const float* __restrict__ pf, const float* __restrict__ pd,
    const float* __restrict__ vshaped, float* __restrict__ vposed) {
  int b = threadIdx.x;                    // one block of 256
  const float* pfrow = pf + (size_t)b * KP_;
  float acc = vshaped[(size_t)b * N3_ + (N3_ - 1)];
  for (int p = 0; p < P_; ++p)
    acc += pfrow[p] * pd[(size_t)p * N3_ + (N3_ - 1)];
  vposed[(size_t)b * N3_ + (N3_ - 1)] = acc;
}

// ---------------------------------------------------------------------------
// 6) Skinning (WMMA f32 16x16x4): T[v][e] = sum_j W[v][j]*rel[b][j][e],
//    fused vertex transform via LDS tile bounce. One wave per (16 v x 1 b).
// ---------------------------------------------------------------------------
__global__ __launch_bounds__(256) void k_verts(
    const float* __restrict__ W, const float* __restrict__ rel,
    const float* __restrict__ vposed, float* __restrict__ verts) {
  __shared__ float Tl[8][16][17];
  int wslot = threadIdx.x >> 5;
  int wid = blockIdx.x * (blockDim.x >> 5) + wslot;
  int b  = wid / VT_;
  int vt = wid - b * VT_;
  int v0 = vt * 16;
  int lane = threadIdx.x & 31;
  int kp = (lane >> 4) * 2;
  int e  = lane & 15;
  int vload = min(v0 + (lane & 15), V_ - 1);      // clamped row: loads always valid
  const float* wrow = W + (size_t)vload * J_;
  const float* relb = rel + (size_t)b * JP_ * 16;
  v8f acc = (v8f){0,0,0,0,0,0,0,0};

  // main: k = 0..51, unguarded
  for (int k0 = 0; k0 < 52; k0 += 4) {
    int ka = k0 + kp, kb = ka + 1;
    v2f a, bm;
    a.x  = wrow[ka];          a.y  = wrow[kb];
    bm.x = relb[ka * 16 + e]; bm.y = relb[kb * 16 + e];
    acc = __builtin_amdgcn_wmma_f32_16x16x4_f32(false, a, false, bm, (short)0, acc, false, false);
  }
  // epilogue: k = 52..55. rel row 55 is zero-padded, so clamp the W index for
  // kb=55 and let 0 * finite kill the term. Branch-free.
  {
    int ka = 52 + kp, kb = ka + 1;
    v2f a, bm;
    a.x  = wrow[ka];
    a.y  = wrow[min(kb, J_ - 1)];
    bm.x = relb[ka * 16 + e];
    bm.y = relb[kb * 16 + e];                     // row 55 = zeros
    acc = __builtin_amdgcn_wmma_f32_16x16x4_f32(false, a, false, bm, (short)0, acc, false, false);
  }
  int hi = lane >> 4;
#pragma unroll
  for (int r = 0; r < 8; ++r) Tl[wslot][r + hi * 8][e] = acc[r];
  __syncthreads();   // grid is exact (no divergent exits); cross-lane via LDS

  if (lane < 16) {
    int v = v0 + lane;
    if (v < V_) {
      float t[16];
#pragma unroll
      for (int q = 0; q < 16; ++q) t[q] = Tl[wslot][lane][q];
      float px = vposed[(size_t)b * N3_ + v * 3 + 0];
      float py = vposed[(size_t)b * N3_ + v * 3 + 1];
      float pz = vposed[(size_t)b * N3_ + v * 3 + 2];
#pragma unroll
      for (int i = 0; i < 3; ++i) {
        float o = t[i * 4 + 0] * px + t[i * 4 + 1] * py + t[i * 4 + 2] * pz + t[i * 4 + 3];
        verts[(size_t)b * N3_ + v * 3 + i] = o;
      }
    }
  }
}

// ---------------------------------------------------------------------------
extern "C" void kernel_launch(void* const* d_in, const int* in_sizes, int n_in,
                              void* d_out, int out_size, void* d_ws, size_t ws_size,
                              hipStream_t stream) {
  const float* betas      = (const float*)d_in[0];
  const float* pose       = (const float*)d_in[1];
  const float* v_template = (const float*)d_in[2];
  const float* shapedirs  = (const float*)d_in[3];
  const float* posedirs   = (const float*)d_in[4];
  const float* Jreg       = (const float*)d_in[5];
  const float* lbsw       = (const float*)d_in[6];
  const int*   parents    = (const int*)d_in[7];
  float* out = (float*)d_out;
  float* verts  = out;                           // B*V*3
  float* joints = out + (size_t)B_ * N3_;        // B*J*3

  // workspace carve-up (floats); total ~66.4 MB
  float* ws      = (float*)d_ws;
  float* vshaped = ws;
  float* vposed  = vshaped + (size_t)B_ * N3_;
  float* pf      = vposed  + (size_t)B_ * N3_;
  float* rotm    = pf      + (size_t)B_ * KP_;
  float* jloc    = rotm    + (size_t)B_ * J_ * 9;
  float* rel     = jloc    + (size_t)B_ * J_ * 3;   // B * JP_ * 16

  k_shape<<<(B_ * V_ + 255) / 256, 256, 0, stream>>>(betas, v_template, shapedirs, vshaped);
  k_rodrigues<<<(B_ * J_ + 255) / 256, 256, 0, stream>>>(pose, rotm, pf);
  k_jloc<<<(B_ * J_) / 8, 256, 0, stream>>>(Jreg, vshaped, jloc);
  k_chain<<<B_ / 16, 256, 0, stream>>>(rotm, jloc, parents, rel, joints);
  k_pose_gemm<<<(16 * NGF_ + 7) / 8, 256, 0, stream>>>(pf, posedirs, vshaped, vposed);
  k_pose_lastcol<<<1, 256, 0, stream>>>(pf, posedirs, vshaped, vposed);
  k_verts<<<(B_ * VT_) / 8, 256, 0, stream>>>(lbsw, rel, vposed, verts);
}